// ObjectExtractor_27814208209070
// MI455X (gfx1250) — compile-verified
//
#include <hip/hip_runtime.h>
#include <stdint.h>

// Problem constants (match reference).
#define HH      30
#define WW      30
#define HWC     900          // H*W
#define NCOL    10
#define MAXOBJ  16
#define POOLN   5
#define NT      256          // 8 wave32s per workgroup
#define NWAVE   (NT / 32)
#define BIGL    (1 << 20)

// Low 32 bits of a generic (flat) shared pointer are the wave-relative LDS
// byte offset on gfx1250 (aperture base lives in the high bits).
__device__ __forceinline__ uint32_t lds_off_of(const void* p) {
    return (uint32_t)(uintptr_t)p;
}

// Wave32 max-reduction: no barriers, lowered to lane-permute ops.
__device__ __forceinline__ int wave_max_i32(int v) {
    #pragma unroll
    for (int off = 16; off > 0; off >>= 1) {
        const int o = __shfl_xor(v, off, 32);
        v = o > v ? o : v;
    }
    return v;
}

__global__ __launch_bounds__(NT)
void objext_kernel(const int* __restrict__ grids,
                   const int* __restrict__ masks,
                   float* __restrict__ feats_out,   // (B,16,36)
                   float* __restrict__ col_out,     // (B,16)
                   float* __restrict__ val_out)     // (B,16)
{
    __shared__ int   sh_g[HWC];
    __shared__ int   sh_m[HWC];      // mask; later reused as top-k selection keys
    __shared__ int   sh_lab[HWC];
    __shared__ int   sh_size[HWC];
    __shared__ int   sh_rmin[HWC];
    __shared__ int   sh_rmax[HWC];
    __shared__ int   sh_cmin[HWC];
    __shared__ int   sh_cmax[HWC];
    __shared__ int   sh_rsum[HWC];
    __shared__ int   sh_csum[HWC];
    __shared__ int   sh_slot[HWC];
    __shared__ int   sh_red[NWAVE];
    __shared__ int   sh_counts[NCOL];
    __shared__ int   sh_topL[MAXOBJ];
    __shared__ int   sh_valid[MAXOBJ];
    __shared__ float sh_pool[MAXOBJ * POOLN * POOLN];
    __shared__ int   sh_bg;
    __shared__ int   sh_changed;

    const int  b    = blockIdx.x;
    const int  tid  = threadIdx.x;
    const int  lane = tid & 31;
    const int  wave = tid >> 5;
    const long base = (long)b * HWC;

    // ---- Stage grid + mask into LDS via CDNA5 async-to-LDS loads (ASYNCcnt) ----
    for (int i = tid; i < HWC; i += NT) {
        uint32_t lg = lds_off_of(&sh_g[i]);
        uint64_t ag = (uint64_t)(uintptr_t)(grids + base + i);
        asm volatile("global_load_async_to_lds_b32 %0, %1, off"
                     :: "v"(lg), "v"(ag) : "memory");
        uint32_t lm = lds_off_of(&sh_m[i]);
        uint64_t am = (uint64_t)(uintptr_t)(masks + base + i);
        asm volatile("global_load_async_to_lds_b32 %0, %1, off"
                     :: "v"(lm), "v"(am) : "memory");
    }
    asm volatile("s_wait_asynccnt 0" ::: "memory");
    __syncthreads();

    // ---- Background color = argmax of masked color histogram (first max) ----
    if (tid < NCOL) sh_counts[tid] = 0;
    __syncthreads();
    for (int i = tid; i < HWC; i += NT) {
        if (sh_m[i]) atomicAdd(&sh_counts[sh_g[i]], 1);
        sh_size[i] = 0;
        sh_rmin[i] = HH;  sh_rmax[i] = -1;
        sh_cmin[i] = WW;  sh_cmax[i] = -1;
        sh_rsum[i] = 0;   sh_csum[i] = 0;
        sh_slot[i] = MAXOBJ;
    }
    __syncthreads();
    if (tid == 0) {
        int best = -1, bg = 0;
        for (int c = 0; c < NCOL; ++c)
            if (sh_counts[c] > best) { best = sh_counts[c]; bg = c; }
        sh_bg = bg;
    }
    __syncthreads();
    const int bg = sh_bg;

    // ---- Initialize labels: active cells get their own flat index ----
    for (int i = tid; i < HWC; i += NT) {
        const bool act = (sh_m[i] != 0) && (sh_g[i] != bg);
        sh_lab[i] = act ? i : BIGL;
    }
    __syncthreads();

    // ---- CCL: neighbor-min + pointer-jumping compression until stable ----
    for (;;) {
        if (tid == 0) sh_changed = 0;
        __syncthreads();

        for (int i = tid; i < HWC; i += NT) {
            const int l = sh_lab[i];
            if (l < BIGL) {
                const int r = i / WW, c = i % WW, gi = sh_g[i];
                int best = l;
                if (r > 0)      { const int n = i - WW; if (sh_g[n] == gi) { const int ln = sh_lab[n]; if (ln < best) best = ln; } }
                if (r < HH - 1) { const int n = i + WW; if (sh_g[n] == gi) { const int ln = sh_lab[n]; if (ln < best) best = ln; } }
                if (c > 0)      { const int n = i - 1;  if (sh_g[n] == gi) { const int ln = sh_lab[n]; if (ln < best) best = ln; } }
                if (c < WW - 1) { const int n = i + 1;  if (sh_g[n] == gi) { const int ln = sh_lab[n]; if (ln < best) best = ln; } }
                if (best < l) { sh_lab[i] = best; sh_changed = 1; }
            }
        }
        __syncthreads();

        // Path compression: labels only decrease, chains stay in-component.
        for (int i = tid; i < HWC; i += NT) {
            int l = sh_lab[i];
            if (l < BIGL) {
                int p = sh_lab[l];
                while (p < l) { l = p; p = sh_lab[l]; }
                sh_lab[i] = l;
            }
        }
        __syncthreads();
        if (sh_changed == 0) break;   // uniform across the workgroup
        __syncthreads();
    }

    // ---- Per-segment stats via LDS atomics ----
    for (int i = tid; i < HWC; i += NT) {
        const int l = sh_lab[i];
        if (l < BIGL) {
            const int r = i / WW, c = i % WW;
            atomicAdd(&sh_size[l], 1);
            atomicMin(&sh_rmin[l], r);  atomicMax(&sh_rmax[l], r);
            atomicMin(&sh_cmin[l], c);  atomicMax(&sh_cmax[l], c);
            atomicAdd(&sh_rsum[l], r);  atomicAdd(&sh_csum[l], c);
        }
    }
    __syncthreads();

    // ---- Top-16 by size desc, ties -> smaller label (stable argsort(-size)) ----
    for (int i = tid; i < HWC; i += NT)
        sh_m[i] = (sh_size[i] << 10) | (1023 - i);   // size<=900, label<900: fits
    __syncthreads();

    for (int k = 0; k < MAXOBJ; ++k) {
        int lb = -1;
        for (int i = tid; i < HWC; i += NT) { const int key = sh_m[i]; if (key > lb) lb = key; }
        lb = wave_max_i32(lb);                        // shuffle reduce, no barrier
        if (lane == 0) sh_red[wave] = lb;
        __syncthreads();
        if (tid == 0) {
            int best = sh_red[0];
            #pragma unroll
            for (int w = 1; w < NWAVE; ++w) { const int o = sh_red[w]; if (o > best) best = o; }
            const int L = 1023 - (best & 1023);
            sh_topL[k] = L;
            sh_slot[L] = k;
            sh_m[L]    = -1;     // remove from candidate set
        }
        __syncthreads();
    }

    // ---- 11 scalar features + colors + valid (one thread per object slot) ----
    if (tid < MAXOBJ) {
        const int L  = sh_topL[tid];
        const int sz = sh_size[L];
        const int v  = (sz > 0) ? 1 : 0;
        sh_valid[tid] = v;

        float f[11];
        if (v) {
            const float rmin = (float)sh_rmin[L], rmax = (float)sh_rmax[L];
            const float cmin = (float)sh_cmin[L], cmax = (float)sh_cmax[L];
            const float bh = rmax - rmin + 1.0f;
            const float bw = cmax - cmin + 1.0f;
            const float szf = (float)sz;
            const float ss  = fmaxf(szf, 1.0f);
            const float cy  = (float)sh_rsum[L] / ss;
            const float cx  = (float)sh_csum[L] / ss;
            const float dH1 = (float)(HH - 1), dW1 = (float)(WW - 1);
            f[0]  = rmin / dH1;  f[1]  = cmin / dW1;
            f[2]  = rmax / dH1;  f[3]  = cmax / dW1;
            f[4]  = bh / (float)HH;  f[5] = bw / (float)WW;
            f[6]  = cy / dH1;    f[7]  = cx / dW1;
            f[8]  = szf / (float)(HH * WW);
            f[9]  = szf / fmaxf(bh * bw, 1.0f);
            f[10] = bw  / fmaxf(bh, 1.0f);
        } else {
            #pragma unroll
            for (int j = 0; j < 11; ++j) f[j] = 0.0f;
        }
        float* fb = feats_out + ((long)b * MAXOBJ + tid) * 36;
        #pragma unroll
        for (int j = 0; j < 11; ++j) fb[j] = f[j];
        col_out[(long)b * MAXOBJ + tid] = v ? (float)sh_g[L] : 0.0f;
        val_out[(long)b * MAXOBJ + tid] = v ? 1.0f : 0.0f;
    }
    for (int j = tid; j < MAXOBJ * POOLN * POOLN; j += NT) sh_pool[j] = 0.0f;
    __syncthreads();

    // ---- 5x5 bbox-normalized pooling via ds_add_f32 LDS atomics ----
    for (int i = tid; i < HWC; i += NT) {
        const int l = sh_lab[i];
        if (l < BIGL) {
            const int sl = sh_slot[l];
            if (sl < MAXOBJ) {
                const int r = i / WW, c = i % WW;
                const int r0 = sh_rmin[l], c0 = sh_cmin[l];
                int bh = sh_rmax[l] - r0 + 1;  bh = bh < 1 ? 1 : (bh > HH ? HH : bh);
                int bw = sh_cmax[l] - c0 + 1;  bw = bw < 1 ? 1 : (bw > WW ? WW : bw);
                const int lr = r - r0, lc = c - c0;
                float wr[POOLN], wc[POOLN];
                #pragma unroll
                for (int p = 0; p < POOLN; ++p) {
                    int st = (p * bh) / POOLN;
                    int en = ((p + 1) * bh + (POOLN - 1)) / POOLN;
                    int dn = en - st; if (dn < 1) dn = 1;
                    wr[p] = (lr >= st && lr < en) ? (1.0f / (float)dn) : 0.0f;
                    st = (p * bw) / POOLN;
                    en = ((p + 1) * bw + (POOLN - 1)) / POOLN;
                    dn = en - st; if (dn < 1) dn = 1;
                    wc[p] = (lc >= st && lc < en) ? (1.0f / (float)dn) : 0.0f;
                }
                #pragma unroll
                for (int pr = 0; pr < POOLN; ++pr) {
                    if (wr[pr] != 0.0f) {
                        #pragma unroll
                        for (int pc = 0; pc < POOLN; ++pc) {
                            if (wc[pc] != 0.0f)
                                atomicAdd(&sh_pool[(sl * POOLN + pr) * POOLN + pc],
                                          wr[pr] * wc[pc]);
                        }
                    }
                }
            }
        }
    }
    __syncthreads();

    // ---- Write pooled features (zeroed for invalid slots) ----
    for (int j = tid; j < MAXOBJ * POOLN * POOLN; j += NT) {
        const int sl = j / (POOLN * POOLN);
        const int pj = j % (POOLN * POOLN);
        const float val = sh_valid[sl] ? sh_pool[j] : 0.0f;
        feats_out[((long)b * MAXOBJ + sl) * 36 + 11 + pj] = val;
    }
}

extern "C" void kernel_launch(void* const* d_in, const int* in_sizes, int n_in,
                              void* d_out, int out_size, void* d_ws, size_t ws_size,
                              hipStream_t stream) {
    const int* grids = (const int*)d_in[0];   // (B,30,30) int32
    const int* masks = (const int*)d_in[1];   // (B,30,30) int32 (bool)
    const int  B     = in_sizes[0] / HWC;

    float* out   = (float*)d_out;
    float* feats = out;                                   // B*16*36
    float* cols  = out + (size_t)B * MAXOBJ * 36;         // B*16
    float* vals  = cols + (size_t)B * MAXOBJ;             // B*16

    objext_kernel<<<dim3(B), dim3(NT), 0, stream>>>(grids, masks, feats, cols, vals);
}